// GNN_81441169866891
// MI455X (gfx1250) — compile-verified
//
#include <hip/hip_runtime.h>
#include <stdint.h>

typedef __attribute__((ext_vector_type(2))) float v2f;
typedef __attribute__((ext_vector_type(8))) float v8f;

#define HID 64

// ---------------------------------------------------------------------------
// Zero the atomic accumulators (agg1 [N*2], deg [N], agg2 [N*64] contiguous).
// ---------------------------------------------------------------------------
__global__ void gnn_zero_kernel(float* __restrict__ p, long long n) {
    long long i = (long long)blockIdx.x * blockDim.x + threadIdx.x;
    if (i < n) p[i] = 0.0f;
}

// ---------------------------------------------------------------------------
// Layer-1 edge scatter: agg1[dst] += x[src] (2 ch), deg[dst] += 1.
// ---------------------------------------------------------------------------
__global__ void gnn_scatter1_kernel(const float* __restrict__ x,
                                    const long long* __restrict__ src,
                                    const long long* __restrict__ dst,
                                    float* __restrict__ agg1,
                                    float* __restrict__ deg, int E) {
    int e = blockIdx.x * blockDim.x + threadIdx.x;
    if (e >= E) return;
    int s = (int)src[e];
    int d = (int)dst[e];
    atomicAdd(&agg1[d * 2 + 0], x[s * 2 + 0]);
    atomicAdd(&agg1[d * 2 + 1], x[s * 2 + 1]);
    atomicAdd(&deg[d], 1.0f);
}

// ---------------------------------------------------------------------------
// Layer 1: h = relu( (agg1/deg) @ W1l.T + b1l + x @ W1r.T ).  K=2 -> scalar FMAs.
// One thread per (node, channel).
// ---------------------------------------------------------------------------
__global__ void gnn_layer1_kernel(const float* __restrict__ x,
                                  const float* __restrict__ agg1,
                                  const float* __restrict__ deg,
                                  const float* __restrict__ W1l,
                                  const float* __restrict__ b1l,
                                  const float* __restrict__ W1r,
                                  float* __restrict__ h, int N) {
    int idx = blockIdx.x * blockDim.x + threadIdx.x;
    if (idx >= N * HID) return;
    int i = idx >> 6;
    int c = idx & 63;
    float dinv = 1.0f / fmaxf(deg[i], 1.0f);
    float m0 = agg1[i * 2 + 0] * dinv;
    float m1 = agg1[i * 2 + 1] * dinv;
    float x0 = x[i * 2 + 0];
    float x1 = x[i * 2 + 1];
    float v = fmaf(m0, W1l[c * 2 + 0],
              fmaf(m1, W1l[c * 2 + 1],
              fmaf(x0, W1r[c * 2 + 0],
              fmaf(x1, W1r[c * 2 + 1], b1l[c]))));
    h[idx] = fmaxf(v, 0.0f);
}

// ---------------------------------------------------------------------------
// Layer-2 edge scatter: agg2[dst][0:64] += h[src][0:64].
// 16 threads per edge, float4-wide reads, 4 f32 atomics each.
// h and agg2 are L2-resident (25.6 MB each vs 192 MB L2).
// ---------------------------------------------------------------------------
__global__ void gnn_scatter2_kernel(const float* __restrict__ h,
                                    const long long* __restrict__ src,
                                    const long long* __restrict__ dst,
                                    float* __restrict__ agg2, int E) {
    long long t = (long long)blockIdx.x * blockDim.x + threadIdx.x;
    long long e = t >> 4;
    if (e >= E) return;
    int c4 = ((int)t & 15) * 4;
    int s = (int)src[e];
    int d = (int)dst[e];
    const float4 v = *(const float4*)(h + (long long)s * HID + c4);
    float* o = agg2 + (long long)d * HID + c4;
    atomicAdd(o + 0, v.x);
    atomicAdd(o + 1, v.y);
    atomicAdd(o + 2, v.z);
    atomicAdd(o + 3, v.w);
}

// ---------------------------------------------------------------------------
// Layer 2 via V_WMMA_F32_16X16X4_F32 (full fp32, matches reference precision):
//   z = (agg2/deg) @ W2l.T + b2l + h @ W2r.T
// Wave-per-16x16-C-tile. Block = 256 threads = 8 waves = 2 node-tiles x 4 n-tiles.
// No divergence: grid covers exactly (N/32) blocks, so EXEC is all-ones for WMMA.
//
// 32-bit A 16x4 layout: lane l holds row (l&15); v0 = K=2*(l>>4), v1 = K+1.
// 32-bit B 4x16: lane l holds col (l&15); v0 = K=2*(l>>4), v1 = K+1.
// C/D: VGPR r holds M = 8*(l>>4)+r, N = (l&15).
// ---------------------------------------------------------------------------
__global__ void gnn_layer2_wmma_kernel(const float* __restrict__ agg2,
                                       const float* __restrict__ deg,
                                       const float* __restrict__ h,
                                       const float* __restrict__ W2l,
                                       const float* __restrict__ b2l,
                                       const float* __restrict__ W2r,
                                       float* __restrict__ z) {
    const int wave = threadIdx.x >> 5;          // 0..7
    const int lane = threadIdx.x & 31;
    const int mtile = blockIdx.x * 2 + (wave >> 2);   // 16-node tile
    const int ntile = wave & 3;                       // 16-col tile
    const int row = mtile * 16 + (lane & 15);         // A-row this lane feeds
    const int kk  = (lane >> 4) * 2;                  // K sub-offset 0 or 2
    const int col = ntile * 16 + (lane & 15);         // B/C column this lane feeds

    const float invdeg = 1.0f / fmaxf(deg[row], 1.0f);
    const float* hrow = h    + (long long)row * HID;
    const float* arow = agg2 + (long long)row * HID;
    const float* wr   = W2r  + (long long)col * HID;  // B[k][n] = W[n][k]
    const float* wl   = W2l  + (long long)col * HID;

    v8f acc = {};
    // h @ W2r.T
    #pragma unroll
    for (int k0 = 0; k0 < HID; k0 += 4) {
        v2f a; a.x = hrow[k0 + kk]; a.y = hrow[k0 + kk + 1];
        v2f b; b.x = wr[k0 + kk];   b.y = wr[k0 + kk + 1];
        acc = __builtin_amdgcn_wmma_f32_16x16x4_f32(
            false, a, false, b, (short)0, acc, false, false);
    }
    // (agg2/deg) @ W2l.T
    #pragma unroll
    for (int k0 = 0; k0 < HID; k0 += 4) {
        v2f a; a.x = arow[k0 + kk] * invdeg; a.y = arow[k0 + kk + 1] * invdeg;
        v2f b; b.x = wl[k0 + kk];            b.y = wl[k0 + kk + 1];
        acc = __builtin_amdgcn_wmma_f32_16x16x4_f32(
            false, a, false, b, (short)0, acc, false, false);
    }

    const float bias = b2l[col];
    const int mbase = mtile * 16 + 8 * (lane >> 4);
    #pragma unroll
    for (int r = 0; r < 8; ++r) {
        z[(long long)(mbase + r) * HID + col] = acc[r] + bias;
    }
}

// Scalar fallback for any node remainder (N % 32 != 0) — not used for N=100000.
__global__ void gnn_layer2_tail_kernel(const float* __restrict__ agg2,
                                       const float* __restrict__ deg,
                                       const float* __restrict__ h,
                                       const float* __restrict__ W2l,
                                       const float* __restrict__ b2l,
                                       const float* __restrict__ W2r,
                                       float* __restrict__ z, int base, int N) {
    int idx = blockIdx.x * blockDim.x + threadIdx.x;
    int i = base + (idx >> 6);
    int c = idx & 63;
    if (i >= N) return;
    float dinv = 1.0f / fmaxf(deg[i], 1.0f);
    float acc = b2l[c];
    for (int k = 0; k < HID; ++k) {
        acc = fmaf(agg2[(long long)i * HID + k] * dinv, W2l[c * HID + k], acc);
        acc = fmaf(h[(long long)i * HID + k], W2r[c * HID + k], acc);
    }
    z[(long long)i * HID + c] = acc;
}

// ---------------------------------------------------------------------------
// Decode: scores[e] = dot(z[row[e]], z[col[e]]).  16 lanes per pair, float4
// loads (z is L2-resident), shuffle tree reduction within the 16-lane group.
// ---------------------------------------------------------------------------
__global__ void gnn_decode_kernel(const float* __restrict__ z,
                                  const long long* __restrict__ row,
                                  const long long* __restrict__ col,
                                  float* __restrict__ out, int L) {
    long long t = (long long)blockIdx.x * blockDim.x + threadIdx.x;
    long long e = t >> 4;
    if (e >= L) return;
    int g = (int)t & 15;
    int r = (int)row[e];
    int c = (int)col[e];
    float4 a = *(const float4*)(z + (long long)r * HID + g * 4);
    float4 b = *(const float4*)(z + (long long)c * HID + g * 4);
    float s = a.x * b.x + a.y * b.y + a.z * b.z + a.w * b.w;
    s += __shfl_down(s, 8, 16);
    s += __shfl_down(s, 4, 16);
    s += __shfl_down(s, 2, 16);
    s += __shfl_down(s, 1, 16);
    if (g == 0) out[e] = s;
}

// ---------------------------------------------------------------------------
// Inputs (setup_inputs order):
//   0: x [N,2] f32      1: edge_index [2,E] i64   2: edge_label_index [2,L] i64
//   3: W1l [64,2]       4: b1l [64]               5: W1r [64,2]
//   6: W2l [64,64]      7: b2l [64]               8: W2r [64,64]
// Output: scores [L] f32.
// ---------------------------------------------------------------------------
extern "C" void kernel_launch(void* const* d_in, const int* in_sizes, int n_in,
                              void* d_out, int out_size, void* d_ws, size_t ws_size,
                              hipStream_t stream) {
    const float*     x   = (const float*)d_in[0];
    const long long* ei  = (const long long*)d_in[1];
    const long long* eli = (const long long*)d_in[2];
    const float*     W1l = (const float*)d_in[3];
    const float*     b1l = (const float*)d_in[4];
    const float*     W1r = (const float*)d_in[5];
    const float*     W2l = (const float*)d_in[6];
    const float*     b2l = (const float*)d_in[7];
    const float*     W2r = (const float*)d_in[8];
    float* out = (float*)d_out;

    const int N = in_sizes[0] / 2;
    const int E = in_sizes[1] / 2;
    const int L = in_sizes[2] / 2;

    const long long* src = ei;
    const long long* dst = ei + E;
    const long long* lrow = eli;
    const long long* lcol = eli + L;

    // Workspace layout: [agg1 N*2 | deg N | agg2 N*64 | h N*64 | z N*64]
    float* agg1 = (float*)d_ws;
    float* deg  = agg1 + (long long)N * 2;
    float* agg2 = deg  + N;
    float* h    = agg2 + (long long)N * HID;
    float* z    = h    + (long long)N * HID;

    const int B = 256;

    // 1) zero accumulators (agg1, deg, agg2 are contiguous): N*(2+1+64) floats
    long long nz = (long long)N * (2 + 1 + HID);
    gnn_zero_kernel<<<(unsigned)((nz + B - 1) / B), B, 0, stream>>>(agg1, nz);

    // 2) layer-1 edge scatter
    gnn_scatter1_kernel<<<(E + B - 1) / B, B, 0, stream>>>(x, src, dst, agg1, deg, E);

    // 3) layer-1 dense + relu
    gnn_layer1_kernel<<<((long long)N * HID + B - 1) / B, B, 0, stream>>>(
        x, agg1, deg, W1l, b1l, W1r, h, N);

    // 4) layer-2 edge scatter (16 threads/edge)
    long long t2 = (long long)E * 16;
    gnn_scatter2_kernel<<<(unsigned)((t2 + B - 1) / B), B, 0, stream>>>(h, src, dst, agg2, E);

    // 5) layer-2 dense via WMMA (32 nodes per block); scalar tail for remainder
    int full = N / 32;
    if (full > 0)
        gnn_layer2_wmma_kernel<<<full, B, 0, stream>>>(agg2, deg, h, W2l, b2l, W2r, z);
    int rem = N - full * 32;
    if (rem > 0)
        gnn_layer2_tail_kernel<<<(rem * HID + B - 1) / B, B, 0, stream>>>(
            agg2, deg, h, W2l, b2l, W2r, z, full * 32, N);

    // 6) decode (16 threads/pair)
    long long t3 = (long long)L * 16;
    gnn_decode_kernel<<<(unsigned)((t3 + B - 1) / B), B, 0, stream>>>(z, lrow, lcol, out, L);
}